// SpectralSSM_65077344469103
// MI455X (gfx1250) — compile-verified
//
#include <hip/hip_runtime.h>

typedef float v2f __attribute__((ext_vector_type(2)));
typedef float v8f __attribute__((ext_vector_type(8)));

#define N_TOT 4096
#define CB    224
#define DM    64
#define DS    16
#define LN_EPS 1e-5f

// 512 threads = 16 wave32's. Each block: 16 rows of n, all 64 d, all 16 s.
// Each wave: 4 d-tiles (16 n x 16 s each), FP32 WMMA over K = 224 (56 x k=4).
__global__ __launch_bounds__(512)
void spectral_ssm_kernel(const float* __restrict__ x,
                         const float* __restrict__ w_in,
                         const float* __restrict__ b_in,
                         const float* __restrict__ A_log,
                         const float* __restrict__ Wb,
                         const float* __restrict__ Wc,
                         const float* __restrict__ Dv,
                         const float* __restrict__ Wo,
                         const float* __restrict__ bo,
                         const float* __restrict__ gamma,
                         const float* __restrict__ beta,
                         float* __restrict__ out)
{
    __shared__ float xs[16 * 225];      // x tile, padded row stride (225 mod 64 banks = odd-ish)
    __shared__ float aexp[DM * DS];     // exp(-exp(A_log))
    __shared__ float outs[16 * DM];     // per-n pre-Wo activations
    __shared__ float wbS[DS], bbS[DS], wcS[DS], bcS[DS];

    const int tid  = threadIdx.x;
    const int lane = tid & 31;
    const int wave = tid >> 5;
    const int n0   = blockIdx.x * 16;

    // Stage the 16x224 x-tile into LDS (7 elements per thread)
    for (int e = tid; e < 16 * CB; e += 512) {
        int m = e / CB, c = e - m * CB;
        xs[m * 225 + c] = x[(n0 + m) * CB + c];
    }
    // Decay factors A = exp(-exp(A_log)) in (0,1)
    for (int e = tid; e < DM * DS; e += 512)
        aexp[e] = __expf(-__expf(A_log[e]));
    // Collapsed B/C projections: wb[s] = w_in . Wb[:,s], etc.
    if (tid < DS) {
        float wb = 0.f, bb = 0.f, wc = 0.f, bc = 0.f;
        for (int d2 = 0; d2 < DM; ++d2) {
            float wv = w_in[d2], bv = b_in[d2];
            wb += wv * Wb[d2 * DS + tid];
            bb += bv * Wb[d2 * DS + tid];
            wc += wv * Wc[d2 * DS + tid];
            bc += bv * Wc[d2 * DS + tid];
        }
        wbS[tid] = wb; bbS[tid] = bb; wcS[tid] = wc; bcS[tid] = bc;
    }
    __syncthreads();

    const int s    = lane & 15;       // ds-column within tile (N of WMMA)
    const int hi   = lane >> 4;       // lane half
    const int krow = hi * 2;          // K rows handled by this half
    const int m    = lane & 15;       // A-matrix row (n within tile)

    for (int t = 0; t < 4; ++t) {
        const int d = wave * 4 + t;
        const float A  = aexp[d * DS + s];
        const float A2 = A * A, A4 = A2 * A2;
        // B-fragment = geometric progression of A-powers, advanced by A^4 per k-step.
        // k-step i covers c = 220-4i .. 223-4i; exponent e = 223-c.
        // low half (krow=0): VGPR0 -> A^(3+4i), VGPR1 -> A^(2+4i)
        // high half (krow=2): VGPR0 -> A^(1+4i), VGPR1 -> A^(0+4i)
        float p1 = hi ? 1.0f : A2;
        float p0 = p1 * A;

        v8f acc1 = {};   // P1 = sum_c x   * A^(223-c)
        v8f acc2 = {};   // P2 = sum_c x^2 * A^(223-c)
        for (int i = 0; i < 56; ++i) {
            const int c0 = 220 - 4 * i + krow;
            v2f a, a2, b;
            a.x  = xs[m * 225 + c0];
            a.y  = xs[m * 225 + c0 + 1];
            a2.x = a.x * a.x;
            a2.y = a.y * a.y;
            b.x  = p0;
            b.y  = p1;
            acc1 = __builtin_amdgcn_wmma_f32_16x16x4_f32(false, a,  false, b,
                                                         (short)0, acc1, false, false);
            acc2 = __builtin_amdgcn_wmma_f32_16x16x4_f32(false, a2, false, b,
                                                         (short)0, acc2, false, false);
            p0 *= A4;
            p1 *= A4;
        }

        // Combine: h = w_d*wb_s*P2 + (w_d*bb_s + b_d*wb_s)*P1 + b_d*bb_s*P0geo
        const float w_d = w_in[d], b_d = b_in[d], Dd = Dv[d];
        const float wbs = wbS[s], bbs = bbS[s], wcs = wcS[s], bcs = bcS[s];
        const float A224 = __powf(A, 224.0f);
        const float P0g  = (1.0f - A224) / (1.0f - A);   // A < 1 strictly
        const float coefA = w_d * wbs;
        const float coefB = w_d * bbs + b_d * wbs;
        const float coefC = b_d * bbs * P0g;

        // y[n,d] = sum_s h[n,d,s] * (x_last[n]*wc_s + bc_s); reduce across 16 lanes
        #pragma unroll
        for (int r = 0; r < 8; ++r) {
            const int mrow = r + hi * 8;                 // C/D layout: lanes16-31 hold M=8+r
            const float xlast = xs[mrow * 225 + 223];
            float h    = coefA * acc2[r] + coefB * acc1[r] + coefC;
            float part = h * (xlast * wcs + bcs);
            part += __shfl_xor(part, 1, 32);
            part += __shfl_xor(part, 2, 32);
            part += __shfl_xor(part, 4, 32);
            part += __shfl_xor(part, 8, 32);
            if (s == 0) {
                outs[mrow * DM + d] = part + Dd * (xlast * w_d + b_d);
            }
        }
    }
    __syncthreads();

    // Final stage: wave w owns n = n0 + w. Each lane computes z for d = lane, lane+32,
    // then LayerNorm via wave32 reductions.
    {
        const int n  = n0 + wave;
        const int d1 = lane, d2 = lane + 32;
        float z1 = bo[d1], z2 = bo[d2];
        for (int k = 0; k < DM; ++k) {
            const float ok = outs[wave * DM + k];        // LDS broadcast
            z1 += ok * Wo[d1 * DM + k];
            z2 += ok * Wo[d2 * DM + k];
        }
        float sum = z1 + z2;
        sum += __shfl_xor(sum, 1, 32);
        sum += __shfl_xor(sum, 2, 32);
        sum += __shfl_xor(sum, 4, 32);
        sum += __shfl_xor(sum, 8, 32);
        sum += __shfl_xor(sum, 16, 32);
        const float mu = sum * (1.0f / 64.0f);
        float e1 = z1 - mu, e2 = z2 - mu;
        float sq = e1 * e1 + e2 * e2;
        sq += __shfl_xor(sq, 1, 32);
        sq += __shfl_xor(sq, 2, 32);
        sq += __shfl_xor(sq, 4, 32);
        sq += __shfl_xor(sq, 8, 32);
        sq += __shfl_xor(sq, 16, 32);
        const float rs = rsqrtf(sq * (1.0f / 64.0f) + LN_EPS);
        out[n * DM + d1] = gamma[d1] * e1 * rs + beta[d1];
        out[n * DM + d2] = gamma[d2] * e2 * rs + beta[d2];
    }
}

extern "C" void kernel_launch(void* const* d_in, const int* in_sizes, int n_in,
                              void* d_out, int out_size, void* d_ws, size_t ws_size,
                              hipStream_t stream) {
    (void)in_sizes; (void)n_in; (void)out_size; (void)d_ws; (void)ws_size;
    const float* x     = (const float*)d_in[0];
    const float* w_in  = (const float*)d_in[1];
    const float* b_in  = (const float*)d_in[2];
    const float* A_log = (const float*)d_in[3];
    const float* Wb    = (const float*)d_in[4];
    const float* Wc    = (const float*)d_in[5];
    const float* Dv    = (const float*)d_in[6];
    const float* Wo    = (const float*)d_in[7];
    const float* bo    = (const float*)d_in[8];
    const float* gamma = (const float*)d_in[9];
    const float* beta  = (const float*)d_in[10];
    float* outp = (float*)d_out;

    dim3 grid(N_TOT / 16), block(512);
    hipLaunchKernelGGL(spectral_ssm_kernel, grid, block, 0, stream,
                       x, w_in, b_in, A_log, Wb, Wc, Dv, Wo, bo, gamma, beta, outp);
}